// SMAL_86706799771657
// MI455X (gfx1250) — compile-verified
//
#include <hip/hip_runtime.h>
#include <math.h>

#define V 3889
#define NJ 35
#define NB 30
#define NC 11667   // 3*V
#define KTOT 336   // 30 + 306
#define BATCH 1024
#define NVT 244    // ceil(V/16)

typedef float v2f __attribute__((ext_vector_type(2)));
typedef float v8f __attribute__((ext_vector_type(8)));

__constant__ int PAR[NJ] = {0,0,1,2,3,4,5,6,7,8,9,10,6,12,13,14,6,16,17,18,
                            16,20,21,22,16,24,25,26,27,28,29,30,16,32,32};

// ---------------- kernel 0: JT / JS pre-reduction --------------------------
// row 0..29 -> JS[row] = J_reg^T . shapedirs[row];  row 30 -> JT = J_reg^T . v_template
__global__ __launch_bounds__(64) void k0_jreg(const float* __restrict__ vt,
                                              const float* __restrict__ sd,
                                              const float* __restrict__ Jreg,
                                              float* __restrict__ JT,
                                              float* __restrict__ JS) {
    int bid = blockIdx.x;            // 0..3254
    int row = bid / 105;             // 0..30
    int jc  = bid % 105;
    int j = jc / 3, c = jc % 3;
    int t = threadIdx.x;
    float acc = 0.f;
    for (int v = t; v < V; v += 64) {
        float src = (row < NB) ? sd[row * NC + v * 3 + c] : vt[v * 3 + c];
        acc += src * Jreg[v * NJ + j];
    }
    __shared__ float red[64];
    red[t] = acc;
    __syncthreads();
    for (int o = 32; o > 0; o >>= 1) {
        if (t < o) red[t] += red[t + o];
        __syncthreads();
    }
    if (t == 0) {
        if (row < NB) JS[row * 105 + jc] = red[0];
        else          JT[jc] = red[0];
    }
}

// ---------------- concat [shapedirs; posedirs] -----------------------------
__global__ __launch_bounds__(256) void k_cat(const float* __restrict__ sd,
                                             const float* __restrict__ pd,
                                             float* __restrict__ dirs) {
    long idx = (long)blockIdx.x * 256 + threadIdx.x;
    if (idx < (long)KTOT * NC) {
        int k = (int)(idx / NC), n = (int)(idx % NC);
        dirs[idx] = (k < NB) ? sd[k * NC + n] : pd[(k - NB) * NC + n];
    }
}

// ---------------- per-batch prep: rodrigues, chain, A, pose feature --------
__device__ inline float log_scale(int j, int a, const float* bl) {
    bool g1 = (j==7||j==8||j==9||j==10||j==17||j==18||j==19||j==20);
    bool g2 = (j>=25 && j<=31);
    bool g3 = (j==33||j==34);
    float s = 0.f;
    if (g1) s = (a==2) ? bl[0] : bl[1];
    if (g2) s = (a==0) ? bl[2] : bl[3];
    if (g3) { if (a==1) s = bl[4]; else if (a==2) s = bl[5]; }
    if (j==16 && a==0) s = bl[6];
    return s;
}

__global__ __launch_bounds__(64) void k1_prep(const float* __restrict__ beta,
                                              const float* __restrict__ bl_in,
                                              const float* __restrict__ theta,
                                              const float* __restrict__ JT,
                                              const float* __restrict__ JS,
                                              float* __restrict__ Aw,
                                              float* __restrict__ pfcat,
                                              float* __restrict__ outRs) {
    int b = blockIdx.x;
    int t = threadIdx.x;
    __shared__ float sR[NJ][9];
    __shared__ float sS[NJ][3];
    __shared__ float sJ[NJ][3];
    __shared__ float sL[NJ][16];
    __shared__ float sW[NJ][16];

    const float* th  = theta + b * NJ * 3;
    const float* bet = beta  + b * NB;
    const float* bl  = bl_in + b * 7;

    if (t < NJ) {
        float tx = th[t*3+0], ty = th[t*3+1], tz = th[t*3+2];
        float ax = tx + 1e-8f, ay = ty + 1e-8f, az = tz + 1e-8f;
        float ang = sqrtf(ax*ax + ay*ay + az*az);
        float rx = tx/ang, ry = ty/ang, rz = tz/ang;
        float c = cosf(ang), s = sinf(ang), mc = 1.f - c;
        float R[9] = { c + mc*rx*rx,      mc*rx*ry - s*rz,  mc*rx*rz + s*ry,
                       mc*ry*rx + s*rz,   c + mc*ry*ry,     mc*ry*rz - s*rx,
                       mc*rz*rx - s*ry,   mc*rz*ry + s*rx,  c + mc*rz*rz };
        for (int i = 0; i < 9; ++i) {
            sR[t][i] = R[i];
            outRs[b * (NJ*9) + t*9 + i] = R[i];
        }
        for (int a = 0; a < 3; ++a) sS[t][a] = expf(log_scale(t, a, bl));
        for (int cc = 0; cc < 3; ++cc) {
            float acc = JT[t*3 + cc];
            for (int k = 0; k < NB; ++k) acc += bet[k] * JS[k*105 + t*3 + cc];
            sJ[t][cc] = acc;
        }
    }
    __syncthreads();

    if (t < NJ) {
        float L[16];
        if (t == 0) {
            for (int r = 0; r < 3; ++r) {
                for (int cc = 0; cc < 3; ++cc) L[r*4+cc] = sR[0][r*3+cc];
                L[r*4+3] = sJ[0][r];
            }
        } else {
            int p = PAR[t];
            for (int r = 0; r < 3; ++r) {
                for (int cc = 0; cc < 3; ++cc)
                    L[r*4+cc] = sR[t][r*3+cc] * sS[t][cc] / sS[p][r];
                L[r*4+3] = sJ[t][r] - sJ[p][r];
            }
        }
        L[12] = 0.f; L[13] = 0.f; L[14] = 0.f; L[15] = 1.f;
        for (int i = 0; i < 16; ++i) sL[t][i] = L[i];
    }
    __syncthreads();

    if (t < 16) sW[0][t] = sL[0][t];
    __syncthreads();
    for (int i = 1; i < NJ; ++i) {
        if (t < 16) {
            int r = t >> 2, cc = t & 3;
            int p = PAR[i];
            float acc = 0.f;
            for (int k = 0; k < 4; ++k) acc += sW[p][r*4+k] * sL[i][k*4+cc];
            sW[i][t] = acc;
        }
        __syncthreads();
    }

    if (t < NJ) {
        float out[16];
        for (int i = 0; i < 16; ++i) out[i] = sW[t][i];
        for (int r = 0; r < 4; ++r)
            out[r*4+3] -= sW[t][r*4+0]*sJ[t][0] + sW[t][r*4+1]*sJ[t][1] + sW[t][r*4+2]*sJ[t][2];
        for (int i = 0; i < 16; ++i) Aw[b * (NJ*16) + t*16 + i] = out[i];
        if (t >= 1) {
            for (int i = 0; i < 9; ++i)
                pfcat[b*KTOT + NB + (t-1)*9 + i] =
                    sR[t][i] - ((i==0 || i==4 || i==8) ? 1.f : 0.f);
        }
    }
    if (t < NB) pfcat[b*KTOT + t] = bet[t];
}

// ---------------- big GEMM: v_posed = [beta|pf] @ dirs + v_template --------
// M=1024 (batch), N=11667, K=336.  WMMA f32 16x16x4, 84 K-steps per tile.
__global__ __launch_bounds__(128) void k2_gemm(const float* __restrict__ pfcat,
                                               const float* __restrict__ dirs,
                                               const float* __restrict__ vt,
                                               float* __restrict__ vposed) {
    __shared__ float sA[16 * 340];   // padded stride to avoid bank conflicts
    int mtile = blockIdx.x;
    int tid = threadIdx.x;
    for (int idx = tid; idx < 16 * KTOT; idx += 128) {
        int m = idx / KTOT, k = idx % KTOT;
        sA[m * 340 + k] = pfcat[(mtile*16 + m) * KTOT + k];
    }
    __syncthreads();

    int wave = tid >> 5;
    int lane = tid & 31;
    int l15  = lane & 15;
    int hi   = lane >> 4;              // 0 or 1
    int ntile = blockIdx.y * 4 + wave;
    if (ntile > 729) ntile = 729;      // clamp (no divergent exit; stores dedupe)
    int ncol = ntile * 16 + l15;
    bool nok = (ncol < NC);
    int ncl  = nok ? ncol : 0;

    v8f acc = {0.f,0.f,0.f,0.f,0.f,0.f,0.f,0.f};
    for (int kk = 0; kk < KTOT/4; ++kk) {
        int k0 = kk*4 + 2*hi;
        v2f a, bb;
        a.x = sA[l15 * 340 + k0];
        a.y = sA[l15 * 340 + k0 + 1];
        float b0 = dirs[(long)k0 * NC + ncl];
        float b1 = dirs[(long)(k0+1) * NC + ncl];
        bb.x = nok ? b0 : 0.f;
        bb.y = nok ? b1 : 0.f;
        acc = __builtin_amdgcn_wmma_f32_16x16x4_f32(false, a, false, bb,
                                                    (short)0, acc, false, false);
    }
    float vtn = nok ? vt[ncol] : 0.f;
    for (int r = 0; r < 8; ++r) {
        int row = r + 8*hi;                       // M within tile (C layout)
        int bidx = mtile*16 + row;
        if (nok) vposed[(long)bidx * NC + ncol] = acc[r] + vtn;
    }
}

// ---------------- skinning: T = weights @ A (WMMA), apply to v_posed -------
// one wave per (batch, 16-vertex tile); waves = 1024*244 exactly
__global__ __launch_bounds__(64) void k3_skin(const float* __restrict__ Aw,
                                              const float* __restrict__ weights,
                                              const float* __restrict__ vposed,
                                              const float* __restrict__ trans,
                                              float* __restrict__ verts) {
    __shared__ float sT[2][16 * 17];
    int wave = threadIdx.x >> 5;
    int lane = threadIdx.x & 31;
    int widx = blockIdx.x * 2 + wave;
    int b = widx / NVT;
    int vtile = widx % NVT;
    int vbase = vtile * 16;
    int l15 = lane & 15, hi = lane >> 4;
    int vv = vbase + l15;
    int vcl = (vv < V) ? vv : (V - 1);
    bool vok = (vv < V);

    v8f acc = {0.f,0.f,0.f,0.f,0.f,0.f,0.f,0.f};
    for (int kk = 0; kk < 9; ++kk) {           // K = 36 (35 + 1 zero pad)
        int k0 = kk*4 + 2*hi;
        v2f a, bb;
        a.x = (vok && k0     < NJ) ? weights[vcl*NJ + k0    ] : 0.f;
        a.y = (vok && k0 + 1 < NJ) ? weights[vcl*NJ + k0 + 1] : 0.f;
        bb.x = (k0     < NJ) ? Aw[b*(NJ*16) + k0    *16 + l15] : 0.f;
        bb.y = (k0 + 1 < NJ) ? Aw[b*(NJ*16) + (k0+1)*16 + l15] : 0.f;
        acc = __builtin_amdgcn_wmma_f32_16x16x4_f32(false, a, false, bb,
                                                    (short)0, acc, false, false);
    }
    for (int r = 0; r < 8; ++r)
        sT[wave][(r + 8*hi) * 17 + l15] = acc[r];
    __syncthreads();

    if (lane < 16) {
        int v = vbase + lane;
        if (v < V) {
            const float* tr = &sT[wave][lane * 17];
            float x = vposed[(long)b * NC + v*3 + 0];
            float y = vposed[(long)b * NC + v*3 + 1];
            float z = vposed[(long)b * NC + v*3 + 2];
            for (int c = 0; c < 3; ++c) {
                float o = tr[c*4+0]*x + tr[c*4+1]*y + tr[c*4+2]*z + tr[c*4+3]
                        + trans[b*3 + c];
                verts[(long)b * NC + v*3 + c] = o;
            }
        }
    }
}

// ---------------- final joint regression -----------------------------------
__global__ __launch_bounds__(128) void k4_joints(const float* __restrict__ verts,
                                                 const float* __restrict__ Jreg,
                                                 float* __restrict__ joints) {
    int b = blockIdx.x;
    int t = threadIdx.x;
    __shared__ float sv[128][3];
    int j = t / 3, c = t % 3;     // valid for t < 105
    float acc = 0.f;
    for (int base = 0; base < V; base += 128) {
        int v = base + t;
        if (v < V) {
            sv[t][0] = verts[(long)b * NC + v*3 + 0];
            sv[t][1] = verts[(long)b * NC + v*3 + 1];
            sv[t][2] = verts[(long)b * NC + v*3 + 2];
        } else {
            sv[t][0] = 0.f; sv[t][1] = 0.f; sv[t][2] = 0.f;
        }
        __syncthreads();
        if (t < 105) {
            int lim = (V - base < 128) ? (V - base) : 128;
            for (int i = 0; i < lim; ++i)
                acc += Jreg[(base + i) * NJ + j] * sv[i][c];
        }
        __syncthreads();
    }
    if (t < 105) joints[b * 105 + t] = acc;
}

// ---------------- launch ----------------------------------------------------
extern "C" void kernel_launch(void* const* d_in, const int* in_sizes, int n_in,
                              void* d_out, int out_size, void* d_ws, size_t ws_size,
                              hipStream_t stream) {
    (void)in_sizes; (void)n_in; (void)out_size; (void)ws_size;
    const float* beta        = (const float*)d_in[0];
    const float* betas_limbs = (const float*)d_in[1];
    const float* theta       = (const float*)d_in[2];
    const float* trans       = (const float*)d_in[3];
    const float* v_template  = (const float*)d_in[4];
    const float* shapedirs   = (const float*)d_in[5];
    const float* J_regressor = (const float*)d_in[6];
    const float* posedirs    = (const float*)d_in[7];
    const float* weights     = (const float*)d_in[8];

    float* ws = (float*)d_ws;
    float* Aw     = ws;                 // 1024*35*16            = 573440
    float* pfcat  = ws + 573440;        // 1024*336              = 344064
    float* JT     = ws + 917504;        // 105
    float* JS     = ws + 917632;        // 30*105                = 3150
    float* dirs   = ws + 920832;        // 336*11667             = 3920112
    float* vposed = ws + 4840960;       // 1024*11667            = 11947008

    float* out        = (float*)d_out;
    float* out_verts  = out;                  // 1024*3889*3 = 11947008
    float* out_joints = out + 11948544;       // 1024*35*3   = 107520
    float* out_Rs     = out + 12056064;       // 1024*35*9   = 322560

    k0_jreg<<<3255, 64, 0, stream>>>(v_template, shapedirs, J_regressor, JT, JS);
    k_cat<<<(KTOT * NC + 255) / 256, 256, 0, stream>>>(shapedirs, posedirs, dirs);
    k1_prep<<<BATCH, 64, 0, stream>>>(beta, betas_limbs, theta, JT, JS, Aw, pfcat, out_Rs);
    k2_gemm<<<dim3(64, 183), 128, 0, stream>>>(pfcat, dirs, v_template, vposed);
    k3_skin<<<(BATCH * NVT) / 2, 64, 0, stream>>>(Aw, weights, vposed, trans, out_verts);
    k4_joints<<<BATCH, 128, 0, stream>>>(out_verts, J_regressor, out_joints);
}